// ReservoirSampler_44538810859811
// MI455X (gfx1250) — compile-verified
//
#include <hip/hip_runtime.h>
#include <stdint.h>

// Reservoir-sampler finalize for MI455X (gfx1250).
// No matmul content -> WMMA inapplicable; use the CDNA5 async global<->LDS
// data-mover path (ASYNCcnt) for the gather, wave32 global atomics for the
// segment-max.

#define N_RES    2048
#define CHANNELS 512
#define TPOS     1024   // h*w per batch image (32*32)

typedef __attribute__((address_space(1))) int as1_int;   // global
typedef __attribute__((address_space(3))) int as3_int;   // LDS

// Generic -> global: numeric identity.
__device__ __forceinline__ as1_int* to_global(const void* p) {
    return (as1_int*)(uintptr_t)p;
}
// Generic(shared) -> LDS: flat shared addr is {aperture_hi32, lds_offset32};
// addrspace(3) pointers are the low-32-bit LDS offset, so truncation is correct.
__device__ __forceinline__ as3_int* to_lds(const void* p) {
    return (as3_int*)(uint32_t)(uintptr_t)p;
}

// ---------------------------------------------------------------------------
// Kernel 3 (defined first so its disassembly appears in the snippet):
// one block per reservoir slot; lane ch gathers x[row][ch] via the async
// data mover into LDS, then async-stores the coalesced output row.
// Each lane touches only its own LDS word -> per-wave s_wait_asynccnt is the
// only ordering needed (no workgroup barrier).
// ---------------------------------------------------------------------------
__global__ __launch_bounds__(CHANNELS) void rs_gather_async(
    const float* __restrict__ samples,
    const int*   __restrict__ winner,
    float*       __restrict__ out)
{
    __shared__ float row[CHANNELS];

    const int s  = blockIdx.x;          // reservoir slot
    const int ch = threadIdx.x;         // channel

    const int w = winner[s];            // uniform -> scalar load
    const int r = (w >= 0) ? (N_RES + w) : s;   // source row t in x[t, :]
    const int bi = r >> 10;             // batch index
    const int p  = r & (TPOS - 1);      // h*32+w position

    // x[r][ch] lives at samples[(bi*512 + ch)*1024 + p]
    const float* src = samples + ((size_t)bi * CHANNELS + (size_t)ch) * TPOS + p;

    // Scattered gather: global -> LDS via async data mover (ASYNCcnt path).
    __builtin_amdgcn_global_load_async_to_lds_b32(
        to_global(src), to_lds(&row[ch]), /*offset=*/0, /*cpol=*/0);
    __builtin_amdgcn_s_wait_asynccnt(0);

    // Coalesced write-out: LDS -> global via async data mover.
    float* dst = out + (size_t)s * CHANNELS + ch;
    __builtin_amdgcn_global_store_async_from_lds_b32(
        to_global(dst), to_lds(&row[ch]), /*offset=*/0, /*cpol=*/0);
    __builtin_amdgcn_s_wait_asynccnt(0);
}

// ---------------------------------------------------------------------------
// Kernel 1: winner[s] = -1
// ---------------------------------------------------------------------------
__global__ void rs_init_winner(int* __restrict__ winner)
{
    int i = blockIdx.x * blockDim.x + threadIdx.x;
    if (i < N_RES) winner[i] = -1;
}

// ---------------------------------------------------------------------------
// Kernel 2: segment-max via global atomicMax.
// winner[s] = max{ j : idx[j] == s, s < N_RES }
// ---------------------------------------------------------------------------
__global__ void rs_scatter_max(const int* __restrict__ idx, int M,
                               int* __restrict__ winner)
{
    int j = blockIdx.x * blockDim.x + threadIdx.x;
    if (j < M) {
        int s = idx[j];
        if (s < N_RES) {
            atomicMax(&winner[s], j);   // global_atomic_max_i32, no return
        }
    }
}

extern "C" void kernel_launch(void* const* d_in, const int* in_sizes, int n_in,
                              void* d_out, int out_size, void* d_ws, size_t ws_size,
                              hipStream_t stream)
{
    (void)n_in; (void)out_size; (void)ws_size;

    const float* samples = (const float*)d_in[0];   // (64,512,32,32) f32
    const int*   idx     = (const int*)  d_in[1];   // (M,) i32
    const int    M       = in_sizes[1];             // 63488

    float* out    = (float*)d_out;                  // (2048, 512) f32
    int*   winner = (int*)  d_ws;                   // 2048 * 4 B scratch

    rs_init_winner<<<(N_RES + 255) / 256, 256, 0, stream>>>(winner);
    rs_scatter_max<<<(M + 255) / 256, 256, 0, stream>>>(idx, M, winner);
    rs_gather_async<<<N_RES, CHANNELS, 0, stream>>>(samples, winner, out);
}